// GemoLoss_57286273794866
// MI455X (gfx1250) — compile-verified
//
#include <hip/hip_runtime.h>
#include <hip/hip_bf16.h>
#include <math.h>

// ---------------------------------------------------------------------------
// GemoLoss on MI455X (gfx1250, wave32).
// Bandwidth-bound (~130 MB total traffic @ 23.3 TB/s => ~6-10 us floor).
// Sobel stage implemented as exact-f32 banded matmuls on the WMMA unit:
//   gx(16x16 tile) = Sx(16x18) * Hx(18x16), gy = Dx(16x18) * Hs(18x16)
// via V_WMMA_F32_16X16X4_F32 (5 K-chunks of 4 per product, K padded 18->20).
// Patch staging goes through GLOBAL_LOAD_ASYNC_TO_LDS_B32 (ASYNCcnt) when the
// toolchain exposes the gfx1250 async builtins.
// ---------------------------------------------------------------------------

typedef __attribute__((ext_vector_type(2))) float v2f;
typedef __attribute__((ext_vector_type(8))) float v8f;

#define SIGMA_C 0.03f
#define ALPHA_C 1.0f
#define MASKV  (-1e-08f)

#if defined(__has_builtin)
#if __has_builtin(__builtin_amdgcn_global_load_async_to_lds_b32)
#define HAVE_ASYNC_LDS 1
#endif
#endif

typedef __attribute__((address_space(1))) int gas_i32;
typedef __attribute__((address_space(3))) int las_i32;

// One B32 element: global -> LDS. Async engine when available (per-lane
// addresses, tracked by ASYNCcnt), plain load/store otherwise.
__device__ __forceinline__ void g2lds_b32(const float* __restrict__ g,
                                          float* l) {
#ifdef HAVE_ASYNC_LDS
  __builtin_amdgcn_global_load_async_to_lds_b32(
      (gas_i32*)(void*)g, (las_i32*)(void*)l, 0, 0);
#else
  *l = *g;
#endif
}

__device__ __forceinline__ void wait_async_copies() {
#ifdef HAVE_ASYNC_LDS
#if __has_builtin(__builtin_amdgcn_s_wait_asynccnt)
  __builtin_amdgcn_s_wait_asynccnt(0);
#else
  asm volatile("s_wait_asynccnt 0" ::: "memory");
#endif
#endif
}

// Banded constant B matrices for the horizontal pass.
// Hx[k][n]: -1 at k==n, +1 at k==n+2  (horizontal diff [-1,0,1])
// Hs[k][n]: +1 at k==n, +2 at k==n+1, +1 at k==n+2 (horizontal smooth [1,2,1])
// Both are identically 0 for k >= 18 (n <= 15), so the K=18..19 pad is inert.
__device__ __forceinline__ float hxval(int k, int n) {
  return (k == n) ? -1.f : ((k == n + 2) ? 1.f : 0.f);
}
__device__ __forceinline__ float hsval(int k, int n) {
  return (k == n) ? 1.f : ((k == n + 1) ? 2.f : ((k == n + 2) ? 1.f : 0.f));
}

// ---------------------------------------------------------------------------
// K1: Sobel via WMMA. One wave (32 lanes) per 16x16 output tile.
//   grid = (w/16, h/16, n), block = 32.
// LDS patch has 20 columns (18 data + 2 zero pad) so every A-operand read is
// unconditional -> wave-uniform control flow around the 10 WMMAs (no
// saveexec/cbranch in the inner loop).
// ---------------------------------------------------------------------------
__global__ __launch_bounds__(32) void sobel_wmma_kernel(
    const float* __restrict__ images,   // [n][3][h][w], channel 0 used
    float* __restrict__ edges,          // [n][h*w]
    float* __restrict__ thetas,         // [n][h*w]
    unsigned* __restrict__ emax_bits,   // [n]
    int h, int w) {
  const int c0   = blockIdx.x * 16;
  const int r0   = blockIdx.y * 16;
  const int img  = blockIdx.z;
  const int lane = threadIdx.x;
  const int m    = lane & 15;     // M row (A) / N col (B,C,D)
  const int half = lane >> 4;     // K-half selector

  const float* x = images + (size_t)img * 3 * h * w;  // channel 0

  // 18 rows x 20 cols (cols 18,19 zeroed); row stride 21 to spread banks.
  __shared__ float patch[18][21];
  for (int idx = lane; idx < 18 * 20; idx += 32) {
    int pr = idx / 20, pc = idx % 20;
    if (pc < 18) {
      // Clamped reads only ever feed border outputs, which are masked to
      // zero below (reference zero-pads the VALID conv result).
      int gr = min(max(r0 - 1 + pr, 0), h - 1);
      int gc = min(max(c0 - 1 + pc, 0), w - 1);
      g2lds_b32(&x[(size_t)gr * w + gc], &patch[pr][pc]);
    } else {
      patch[pr][pc] = 0.f;
    }
  }
  wait_async_copies();
  __syncthreads();

  v8f accx = {};  // gx accumulator (16x16 f32 C/D)
  v8f accy = {};  // gy accumulator
#pragma unroll
  for (int q = 0; q < 5; ++q) {
    const int k0 = 4 * q + 2 * half;  // global K for A.x / B reg 0 (<= 18)
    // A operands: vertical pass. Sx = [1,2,1], Dx = [1,0,-1] over patch rows.
    float t0 = patch[m][k0],     u0 = patch[m + 1][k0],     v0 = patch[m + 2][k0];
    float t1 = patch[m][k0 + 1], u1 = patch[m + 1][k0 + 1], v1 = patch[m + 2][k0 + 1];
    v2f Agx = {t0 + 2.f * u0 + v0, t1 + 2.f * u1 + v1};
    v2f Agy = {t0 - v0, t1 - v1};
    // B operands: reg g holds rows (g + 2*half) of the 4-row K chunk.
    v2f Bgx = {hxval(k0, m), hxval(k0 + 1, m)};
    v2f Bgy = {hsval(k0, m), hsval(k0 + 1, m)};
    accx = __builtin_amdgcn_wmma_f32_16x16x4_f32(false, Agx, false, Bgx,
                                                 (short)0, accx, false, false);
    accy = __builtin_amdgcn_wmma_f32_16x16x4_f32(false, Agy, false, Bgy,
                                                 (short)0, accy, false, false);
  }

  float mloc = 0.f;
  size_t ib = (size_t)img * h * w;
#pragma unroll
  for (int g = 0; g < 8; ++g) {
    int i = r0 + g + 8 * half;  // C/D row layout
    int j = c0 + m;
    float gx = accx[g], gy = accy[g];
    bool interior = (i > 0) && (i < h - 1) && (j > 0) && (j < w - 1);
    float e  = interior ? sqrtf(gx * gx + gy * gy) : 0.f;
    float th = interior ? atan2f(gy, gx) : 0.f;
    size_t o = ib + (size_t)i * w + j;
    edges[o]  = e;
    thetas[o] = th;
    mloc = fmaxf(mloc, e);
  }
  for (int off = 16; off > 0; off >>= 1)
    mloc = fmaxf(mloc, __shfl_xor(mloc, off, 32));
  if (lane == 0) atomicMax(&emax_bits[img], __float_as_uint(mloc));
}

// ---------------------------------------------------------------------------
// K2: per-chunk (256 px) counts of edge-mask and valid-mask.
//   grid = (P/256, n), block = 256.
// ---------------------------------------------------------------------------
__global__ void count_kernel(const float* __restrict__ edges,
                             const float* __restrict__ targets,
                             const unsigned* __restrict__ emax_bits,
                             int* __restrict__ cnt_e, int* __restrict__ cnt_v,
                             int P) {
  const int img = blockIdx.y, chunk = blockIdx.x;
  const int p = chunk * 256 + threadIdx.x;
  const size_t base = (size_t)img * P;
  const float thr = 0.1f * __uint_as_float(emax_bits[img]);
  bool pe = edges[base + p] >= thr;
  bool pv = targets[base + p] > MASKV;
  unsigned long long be = __ballot(pe), bv = __ballot(pv);
  __shared__ int se[8], sv[8];
  int wid = threadIdx.x >> 5, lane = threadIdx.x & 31;
  if (lane == 0) { se[wid] = __popcll(be); sv[wid] = __popcll(bv); }
  __syncthreads();
  if (threadIdx.x == 0) {
    int te = 0, tv = 0;
    for (int j = 0; j < 8; ++j) { te += se[j]; tv += sv[j]; }
    size_t cb = (size_t)img * gridDim.x + chunk;
    cnt_e[cb] = te;
    cnt_v[cb] = tv;
  }
}

// ---------------------------------------------------------------------------
// K3: per-image exclusive scan of chunk counts (order-preserving compaction).
//   grid = n, block = 256; nchunks is a multiple of 256.
// ---------------------------------------------------------------------------
__device__ void scan_one(const int* __restrict__ cnt, int* __restrict__ offs,
                         int* __restrict__ total, int nchunks, int tid) {
  __shared__ int tmp[256];
  int base = 0;
  for (int t0 = 0; t0 < nchunks; t0 += 256) {
    int x = cnt[t0 + tid];
    tmp[tid] = x;
    __syncthreads();
    for (int off = 1; off < 256; off <<= 1) {
      int v = (tid >= off) ? tmp[tid - off] : 0;
      __syncthreads();
      tmp[tid] += v;
      __syncthreads();
    }
    offs[t0 + tid] = base + tmp[tid] - x;  // exclusive prefix
    base += tmp[255];
    __syncthreads();
  }
  if (tid == 0) *total = base;
}

__global__ void scan_kernel(const int* cnt_e, const int* cnt_v,
                            int* off_e, int* off_v,
                            int* tot_e, int* tot_v, int nchunks) {
  const int img = blockIdx.x, tid = threadIdx.x;
  const size_t b = (size_t)img * nchunks;
  scan_one(cnt_e + b, off_e + b, tot_e + img, nchunks, tid);
  scan_one(cnt_v + b, off_v + b, tot_v + img, nchunks, tid);
}

// ---------------------------------------------------------------------------
// K4: order-preserving compaction of pixel indices (ascending order ==
// stable argsort of the boolean mask, which is all the reference ever reads).
// ---------------------------------------------------------------------------
__global__ void compact_kernel(const float* __restrict__ edges,
                               const float* __restrict__ targets,
                               const unsigned* __restrict__ emax_bits,
                               const int* __restrict__ off_e,
                               const int* __restrict__ off_v,
                               int* __restrict__ edge_list,
                               int* __restrict__ valid_list, int P) {
  const int img = blockIdx.y, chunk = blockIdx.x;
  const int p = chunk * 256 + threadIdx.x;
  const size_t base = (size_t)img * P;
  const float thr = 0.1f * __uint_as_float(emax_bits[img]);
  bool pe = edges[base + p] >= thr;
  bool pv = targets[base + p] > MASKV;
  unsigned long long be = __ballot(pe), bv = __ballot(pv);
  int wid = threadIdx.x >> 5, lane = threadIdx.x & 31;
  unsigned lmask = (lane == 0) ? 0u : (0xffffffffu >> (32 - lane));
  __shared__ int we[8], wv[8];
  if (lane == 0) { we[wid] = __popcll(be); wv[wid] = __popcll(bv); }
  __syncthreads();
  int pre_e = 0, pre_v = 0;
  for (int j = 0; j < wid; ++j) { pre_e += we[j]; pre_v += wv[j]; }
  const size_t cb = (size_t)img * gridDim.x + chunk;
  if (pe)
    edge_list[base + off_e[cb] + pre_e + __popc((unsigned)be & lmask)] = p;
  if (pv)
    valid_list[base + off_v[cb] + pre_v + __popc((unsigned)bv & lmask)] = p;
}

// ---------------------------------------------------------------------------
// K5: sampling + ranking loss. grid = (ceil(S/256), n), block = 256.
// Random 4-B gathers; each image's inputs/targets (3 MB each) sit in the
// 192 MB L2, so gathers are L2 hits.
// ---------------------------------------------------------------------------
__global__ void loss_kernel(const float* __restrict__ inputs,
                            const float* __restrict__ targets,
                            const float* __restrict__ thetas,
                            const int* __restrict__ edge_list,
                            const int* __restrict__ valid_list,
                            const int* __restrict__ tot_e,
                            const int* __restrict__ tot_v,
                            const int* __restrict__ ra,
                            const int* __restrict__ rd,
                            const int* __restrict__ rp,
                            float* __restrict__ accum,  // [n][2]
                            int S, int h, int w) {
  const int img = blockIdx.y;
  const int s = blockIdx.x * 256 + threadIdx.x;
  const int P = h * w;
  const size_t ib = (size_t)img * P;
  const float* inp = inputs + ib;
  const float* tgt = targets + ib;
  const int ce = tot_e[img], cv = tot_v[img];
  const int minlen = max(ce, 1), nvalid = max(cv, 1);
  float eq = 0.f, uq = 0.f;
  if (s < S) {
    int aidx = ra[(size_t)img * S + s] % minlen;
    // ce==0 => argsort of all-true is identity => index itself
    int anchor = (ce > 0) ? edge_list[ib + aidx] : aidx;
    float th = thetas[ib + anchor];
    int row_a = anchor / w, col_a = anchor % w;
    float ct = cosf(th), st = sinf(th);
    const float sgn[4] = {-1.f, -1.f, 1.f, 1.f};
    int pix[4];
#pragma unroll
    for (int k = 0; k < 4; ++k) {
      float dist = ((float)rd[(size_t)img * 4 * S + (size_t)k * S + s] + 2.0f) * sgn[k];
      int c = col_a + (int)rintf(dist * ct);   // rintf == round-half-even
      int r = row_a + (int)rintf(dist * st);
      c = min(max(c, 0), w - 1);
      r = min(max(r, 0), h - 1);
      pix[k] = r * w + c;
    }
    int sh0 = rp[(size_t)img * 2 * S + 2 * s + 0] % nvalid;
    int sh1 = rp[(size_t)img * 2 * S + 2 * s + 1] % nvalid;
    int rA = (cv > 0) ? valid_list[ib + sh0] : sh0;
    int rB = (cv > 0) ? valid_list[ib + sh1] : sh1;
    int Ai[4] = {pix[0], pix[1], pix[2], rA};
    int Bi[4] = {pix[1], pix[2], pix[3], rB};
    const float hi = 1.0f + SIGMA_C;
    const float lo = 1.0f / (1.0f + SIGMA_C);
#pragma unroll
    for (int k = 0; k < 4; ++k) {
      float iA = inp[Ai[k]], iB = inp[Bi[k]];
      float tA = tgt[Ai[k]], tB = tgt[Bi[k]];
      float cm = ((tA > MASKV) ? 1.f : 0.f) * ((tB > MASKV) ? 1.f : 0.f);
      float ratio = (tA + 1e-6f) / (tB + 1e-6f);
      bool meq = (ratio < hi) && (ratio > lo);
      float lab = ((ratio >= hi) ? 1.f : 0.f) + ((ratio <= lo) ? -1.f : 0.f);
      float d = iA - iB;
      if (meq) eq += d * d * cm;
      else     uq += log1pf(expf((iB - iA) * lab)) * cm;
    }
  }
  // wave32 reduce, then block reduce, then one atomic per block.
  for (int off = 16; off > 0; off >>= 1) {
    eq += __shfl_xor(eq, off, 32);
    uq += __shfl_xor(uq, off, 32);
  }
  __shared__ float re[8], ru[8];
  int wid = threadIdx.x >> 5, lane = threadIdx.x & 31;
  if (lane == 0) { re[wid] = eq; ru[wid] = uq; }
  __syncthreads();
  if (threadIdx.x == 0) {
    float te = 0.f, tu = 0.f;
    for (int j = 0; j < 8; ++j) { te += re[j]; tu += ru[j]; }
    atomicAdd(&accum[img * 2 + 0], te);
    atomicAdd(&accum[img * 2 + 1], tu);
  }
}

__global__ void init_kernel(unsigned* emax_bits, float* accum, int n) {
  int t = blockIdx.x * blockDim.x + threadIdx.x;
  if (t < n) emax_bits[t] = 0u;
  if (t < 2 * n) accum[t] = 0.f;
}

__global__ void finalize_kernel(const float* __restrict__ accum,
                                float* __restrict__ out, int n, float inv4S) {
  if (threadIdx.x == 0 && blockIdx.x == 0) {
    float tot = 0.f;
    for (int i = 0; i < n; ++i)
      tot += (ALPHA_C * accum[2 * i] + accum[2 * i + 1]) * inv4S;
    out[0] = tot / (float)n;
  }
}

// ---------------------------------------------------------------------------
extern "C" void kernel_launch(void* const* d_in, const int* in_sizes, int n_in,
                              void* d_out, int out_size, void* d_ws, size_t ws_size,
                              hipStream_t stream) {
  const int h = 768, w = 1024;
  const int P = h * w;
  const int n = in_sizes[0] / P;            // 4
  const int S = in_sizes[3] / n;            // 131072
  const int nchunks = P / 256;              // 3072

  const float* inputs  = (const float*)d_in[0];
  const float* targets = (const float*)d_in[1];
  const float* images  = (const float*)d_in[2];
  const int*   ra      = (const int*)d_in[3];
  const int*   rd      = (const int*)d_in[4];
  const int*   rp      = (const int*)d_in[5];
  float* out = (float*)d_out;

  // Workspace carve-out (~50.5 MB total for n=4).
  char* p = (char*)d_ws;
  float* edges      = (float*)p;    p += (size_t)n * P * sizeof(float);
  float* thetas     = (float*)p;    p += (size_t)n * P * sizeof(float);
  int*   edge_list  = (int*)p;      p += (size_t)n * P * sizeof(int);
  int*   valid_list = (int*)p;      p += (size_t)n * P * sizeof(int);
  int*   cnt_e      = (int*)p;      p += (size_t)n * nchunks * sizeof(int);
  int*   cnt_v      = (int*)p;      p += (size_t)n * nchunks * sizeof(int);
  int*   off_e      = (int*)p;      p += (size_t)n * nchunks * sizeof(int);
  int*   off_v      = (int*)p;      p += (size_t)n * nchunks * sizeof(int);
  int*   tot_e      = (int*)p;      p += (size_t)n * sizeof(int);
  int*   tot_v      = (int*)p;      p += (size_t)n * sizeof(int);
  unsigned* emaxb   = (unsigned*)p; p += (size_t)n * sizeof(unsigned);
  float* accum      = (float*)p;    p += (size_t)2 * n * sizeof(float);
  (void)ws_size; (void)n_in; (void)out_size;

  init_kernel<<<1, 64, 0, stream>>>(emaxb, accum, n);
  sobel_wmma_kernel<<<dim3(w / 16, h / 16, n), 32, 0, stream>>>(
      images, edges, thetas, emaxb, h, w);
  count_kernel<<<dim3(nchunks, n), 256, 0, stream>>>(
      edges, targets, emaxb, cnt_e, cnt_v, P);
  scan_kernel<<<n, 256, 0, stream>>>(cnt_e, cnt_v, off_e, off_v, tot_e, tot_v,
                                     nchunks);
  compact_kernel<<<dim3(nchunks, n), 256, 0, stream>>>(
      edges, targets, emaxb, off_e, off_v, edge_list, valid_list, P);
  loss_kernel<<<dim3((S + 255) / 256, n), 256, 0, stream>>>(
      inputs, targets, thetas, edge_list, valid_list, tot_e, tot_v, ra, rd, rp,
      accum, S, h, w);
  finalize_kernel<<<1, 32, 0, stream>>>(accum, out, n,
                                        1.0f / (4.0f * (float)S));
}